// MyLayer_27736898797833
// MI455X (gfx1250) — compile-verified
//
#include <hip/hip_runtime.h>

typedef float v4f __attribute__((ext_vector_type(4)));

#define N_COLS 8192
#define N_ROWS 4096
#define NV4    (N_COLS / 4)      // 2048 float4 per row
#define TPB    256
#define COLS4_PER_BLOCK 512      // float4 columns per block (= 2048 floats)
#define ROWS_PER_BLOCK  8

// ---------------------------------------------------------------------------
// Kernel 1: ternary quantization of the 8192-element kernel vector.
// One block: block-wide abs-sum -> threshold -> write q in {-1,0,+1} as f32.
// Wave32 shuffle reduction (gfx1250 ds_swizzle path) + tiny LDS cross-wave.
// ---------------------------------------------------------------------------
__global__ __launch_bounds__(TPB) void ternary_quant_kernel(
    const float* __restrict__ w, float* __restrict__ q) {
  const int tid = threadIdx.x;
  const v4f* __restrict__ w4 = (const v4f*)w;

  v4f vals[8];
  float s = 0.0f;
#pragma unroll
  for (int i = 0; i < 8; ++i) {
    vals[i] = w4[tid + TPB * i];                 // coalesced b128 loads
    s += __builtin_fabsf(vals[i].x) + __builtin_fabsf(vals[i].y) +
         __builtin_fabsf(vals[i].z) + __builtin_fabsf(vals[i].w);
  }

  // wave32 reduction
#pragma unroll
  for (int off = 16; off > 0; off >>= 1) s += __shfl_xor(s, off, 32);

  __shared__ float wave_sum[TPB / 32];
  if ((tid & 31) == 0) wave_sum[tid >> 5] = s;
  __syncthreads();

  float tot = 0.0f;
#pragma unroll
  for (int i = 0; i < TPB / 32; ++i) tot += wave_sum[i];

  const float thr = 0.7f * tot / (float)N_COLS;

  v4f* __restrict__ q4 = (v4f*)q;
#pragma unroll
  for (int i = 0; i < 8; ++i) {
    v4f v = vals[i];
    v4f o;
    // Matches sign(sign(w+t)+sign(w-t)): +1 iff w>=t, -1 iff w<=-t, else 0.
    o.x = (v.x >= thr) ? 1.0f : ((v.x <= -thr) ? -1.0f : 0.0f);
    o.y = (v.y >= thr) ? 1.0f : ((v.y <= -thr) ? -1.0f : 0.0f);
    o.z = (v.z >= thr) ? 1.0f : ((v.z <= -thr) ? -1.0f : 0.0f);
    o.w = (v.w >= thr) ? 1.0f : ((v.w <= -thr) ? -1.0f : 0.0f);
    q4[tid + TPB * i] = o;
  }
}

// ---------------------------------------------------------------------------
// Kernel 2: y = x * q + bias, pure HBM-streaming kernel.
// Block = 2048-column stripe x 8 rows. q/bias float4s held in registers and
// reused across the row loop. x read and y written with non-temporal b128
// accesses (268 MB stream > 192 MB L2 -> NT avoids cache thrash).
// ---------------------------------------------------------------------------
__global__ __launch_bounds__(TPB) void scale_bias_kernel(
    const float* __restrict__ x, const float* __restrict__ q,
    const float* __restrict__ bias, float* __restrict__ y) {
  const int tid = threadIdx.x;
  const int c4a = blockIdx.x * COLS4_PER_BLOCK + tid;   // float4 column index
  const int c4b = c4a + TPB;

  const v4f* __restrict__ q4 = (const v4f*)q;
  const v4f* __restrict__ b4 = (const v4f*)bias;
  const v4f qa = q4[c4a];
  const v4f qb = q4[c4b];
  const v4f ba = b4[c4a];
  const v4f bb = b4[c4b];

  const v4f* __restrict__ x4 = (const v4f*)x;
  v4f* __restrict__ y4 = (v4f*)y;

  const int row0 = blockIdx.y * ROWS_PER_BLOCK;
#pragma unroll
  for (int r = 0; r < ROWS_PER_BLOCK; ++r) {
    const int idx = (row0 + r) * NV4;                   // < 2^23, fits int
    v4f xa = __builtin_nontemporal_load(&x4[idx + c4a]);
    v4f xb = __builtin_nontemporal_load(&x4[idx + c4b]);
    v4f oa = xa * qa + ba;                              // contracts to v_fma
    v4f ob = xb * qb + bb;
    __builtin_nontemporal_store(oa, &y4[idx + c4a]);
    __builtin_nontemporal_store(ob, &y4[idx + c4b]);
  }
}

// ---------------------------------------------------------------------------
// Launch: inputs are (x[4096*8192], kernel[8192], bias[8192]), output f32.
// d_ws holds q (8192 floats = 32 KB).
// ---------------------------------------------------------------------------
extern "C" void kernel_launch(void* const* d_in, const int* in_sizes, int n_in,
                              void* d_out, int out_size, void* d_ws,
                              size_t ws_size, hipStream_t stream) {
  const float* x    = (const float*)d_in[0];
  const float* kern = (const float*)d_in[1];
  const float* bias = (const float*)d_in[2];
  float* out = (float*)d_out;
  float* q   = (float*)d_ws;

  ternary_quant_kernel<<<1, TPB, 0, stream>>>(kern, q);

  dim3 grid(NV4 / COLS4_PER_BLOCK, N_ROWS / ROWS_PER_BLOCK);  // (4, 512)
  scale_bias_kernel<<<grid, TPB, 0, stream>>>(x, q, bias, out);
}